// SpatialHyperedge_14010183320107
// MI455X (gfx1250) — compile-verified
//
#include <hip/hip_runtime.h>
#include <hip/hip_bf16.h>
#include <math.h>

typedef __attribute__((ext_vector_type(2))) float v2f;
typedef __attribute__((ext_vector_type(4))) float f4;
typedef __attribute__((ext_vector_type(8))) float v8f;

#define NN 4096
#define BB 16
#define DD 64
#define LB_TH 0.01f
#define L2_LAMBDA 0.001f
#define RECONS_LAMBDA 0.2f
#define XS 68   // LDS row stride (floats): 272B keeps float4 16B-aligned, staggers banks by 4/row

// ---------------------------------------------------------------------------
// Phase 1: threshold incidence, scatter into W (+ identity on the diagonal),
// per-row l1 (max) and l2 norms. Pure streaming kernel (128 MB of traffic).
// ---------------------------------------------------------------------------
__global__ __launch_bounds__(256) void build_W_kernel(const float* __restrict__ inc,
                                                      float* __restrict__ out_W,
                                                      float* __restrict__ l1,
                                                      float* __restrict__ l2) {
    const int i   = blockIdx.x;
    const int tid = threadIdx.x;
    const float* row  = inc + (size_t)i * (NN - 1);
    float*       wrow = out_W + (size_t)i * NN;

    float lmax = 0.f, lsq = 0.f;
    for (int c = tid; c < NN - 1; c += 256) {
        float v = row[c];
        v = (v > LB_TH) ? v : 0.f;
        wrow[c + (c >= i ? 1 : 0)] = v;
        lmax = fmaxf(lmax, fabsf(v));
        lsq += v * v;
    }
    if (tid == 0) wrow[i] = 1.0f;   // incidence_all = W + eye(N)

    __shared__ float smax[256];
    __shared__ float ssum[256];
    smax[tid] = lmax; ssum[tid] = lsq;
    __syncthreads();
    for (int s = 128; s > 0; s >>= 1) {
        if (tid < s) {
            smax[tid] = fmaxf(smax[tid], smax[tid + s]);
            ssum[tid] += ssum[tid + s];
        }
        __syncthreads();
    }
    if (tid == 0) { l1[i] = smax[0]; l2[i] = sqrtf(ssum[0]); }
}

// ---------------------------------------------------------------------------
// Phase 2: (W+I) @ X via V_WMMA_F32_16X16X4_F32, fused X_proj WMMA (C seeded
// with X so xp = X@r_proj + X), row-norm reduction, final loss combine.
// Block = 128 threads (4 waves). blockIdx.x = 16-row tile, blockIdx.y = batch.
// Each wave owns a 16-wide d-slice of the 64-wide per-batch column block.
// ---------------------------------------------------------------------------
__global__ __launch_bounds__(128) void recon_kernel(const float* __restrict__ X,
                                                    const float* __restrict__ rproj,
                                                    const float* __restrict__ W,   // = W + I (in d_out)
                                                    const float* __restrict__ l1,
                                                    const float* __restrict__ l2,
                                                    float* __restrict__ out_loss) {
    __shared__ float Wlds[16 * XS];
    __shared__ float Xlds[64 * XS];
    __shared__ float Rlds[64 * XS];
    __shared__ float err_lds[16];

    const int tid  = threadIdx.x;
    const int lane = tid & 31;
    const int wv   = tid >> 5;       // wave in block: 0..3
    const int l15  = lane & 15;
    const int hl   = lane >> 4;      // half-wave: 0/1
    const int i0   = blockIdx.x * 16;
    const int b    = blockIdx.y;
    const int d0   = wv * 16;

    const float* Xb = X + (size_t)b * NN * DD;

    // stage r_proj (64x64) into LDS once
    for (int it = 0; it < 8; ++it) {
        int p = tid + it * 128;                 // 1024 float4 slots
        int e = p >> 4, c4 = (p & 15) * 4;
        f4 v = *(const f4*)(rproj + e * DD + c4);
        *(f4*)(&Rlds[e * XS + c4]) = v;
    }
    if (tid < 16) err_lds[tid] = 0.f;

    v8f acc = {0.f, 0.f, 0.f, 0.f, 0.f, 0.f, 0.f, 0.f};

    for (int k0 = 0; k0 < NN; k0 += 64) {
        __syncthreads();   // tiles from previous iteration fully consumed
        // W tile: rows i0..i0+15, cols k0..k0+63  (256 float4)
        for (int it = 0; it < 2; ++it) {
            int p = tid + it * 128;
            int r = p >> 4, c4 = (p & 15) * 4;
            f4 v = *(const f4*)(W + (size_t)(i0 + r) * NN + k0 + c4);
            *(f4*)(&Wlds[r * XS + c4]) = v;
        }
        // X tile: rows k0..k0+63, cols 0..63  (1024 float4)
        for (int it = 0; it < 8; ++it) {
            int p = tid + it * 128;
            int k = p >> 4, c4 = (p & 15) * 4;
            f4 v = *(const f4*)(Xb + (size_t)(k0 + k) * DD + c4);
            *(f4*)(&Xlds[k * XS + c4]) = v;
        }
        __syncthreads();
        #pragma unroll
        for (int kk = 0; kk < 64; kk += 4) {
            const int ka = kk + 2 * hl;
            // A (16x4 f32): lane l15 = row M; v0 = K=ka, v1 = K=ka+1 (halves offset by 2)
            v2f a = *(const v2f*)(&Wlds[l15 * XS + ka]);
            // B (4x16 f32): v0 = row K=ka (cols striped over lanes), v1 = row K=ka+1
            v2f bf;
            bf.x = Xlds[ka * XS + d0 + l15];
            bf.y = Xlds[(ka + 1) * XS + d0 + l15];
            acc = __builtin_amdgcn_wmma_f32_16x16x4_f32(false, a, false, bf,
                                                        (short)0, acc, false, false);
        }
    }
    __syncthreads();

    // restage X rows i0..i0+15 (A matrix for X_proj) into Wlds
    for (int it = 0; it < 2; ++it) {
        int p = tid + it * 128;
        int r = p >> 4, c4 = (p & 15) * 4;
        f4 v = *(const f4*)(Xb + (size_t)(i0 + r) * DD + c4);
        *(f4*)(&Wlds[r * XS + c4]) = v;
    }
    __syncthreads();

    // xp seeded with the raw X tile  ->  after WMMAs xp = X@r_proj + X
    v8f xp;
    #pragma unroll
    for (int v = 0; v < 8; ++v) {
        int m = v + 8 * hl;                  // C/D layout: VGPR v, half hl -> row m
        xp[v] = Wlds[m * XS + d0 + l15];
    }
    #pragma unroll
    for (int kk = 0; kk < 64; kk += 4) {
        const int ka = kk + 2 * hl;
        v2f a = *(const v2f*)(&Wlds[l15 * XS + ka]);
        v2f bf;
        bf.x = Rlds[ka * XS + d0 + l15];
        bf.y = Rlds[(ka + 1) * XS + d0 + l15];
        xp = __builtin_amdgcn_wmma_f32_16x16x4_f32(false, a, false, bf,
                                                   (short)0, xp, false, false);
    }

    // diff = (X_proj + X) - (W+I)@X ; sum diff^2 over d (16 lanes per row here,
    // 4 waves cover the 4 d-slices via LDS float atomics)
    #pragma unroll
    for (int v = 0; v < 8; ++v) {
        float d = xp[v] - acc[v];
        float sq = d * d;
        sq += __shfl_xor(sq, 1, 16);
        sq += __shfl_xor(sq, 2, 16);
        sq += __shfl_xor(sq, 4, 16);
        sq += __shfl_xor(sq, 8, 16);
        if (l15 == 0) atomicAdd(&err_lds[v + 8 * hl], sq);
    }
    __syncthreads();

    if (tid < 16) {
        int row = i0 + tid;
        float loss = RECONS_LAMBDA * sqrtf(err_lds[tid]) + l1[row] + L2_LAMBDA * l2[row];
        out_loss[(size_t)b * NN + row] = loss;
    }
}

// ---------------------------------------------------------------------------
extern "C" void kernel_launch(void* const* d_in, const int* in_sizes, int n_in,
                              void* d_out, int out_size, void* d_ws, size_t ws_size,
                              hipStream_t stream) {
    (void)in_sizes; (void)n_in; (void)out_size; (void)ws_size;
    const float* X     = (const float*)d_in[0];   // [B,N,D]
    const float* rproj = (const float*)d_in[1];   // [D,D]
    const float* inc   = (const float*)d_in[2];   // [N,N-1]

    float* out      = (float*)d_out;
    float* out_loss = out;                        // [B,N]   = 65536 floats
    float* out_W    = out + (size_t)BB * NN;      // [N,N]   = incidence_all

    float* l1 = (float*)d_ws;                     // [N]
    float* l2 = l1 + NN;                          // [N]

    build_W_kernel<<<dim3(NN), dim3(256), 0, stream>>>(inc, out_W, l1, l2);
    recon_kernel<<<dim3(NN / 16, BB), dim3(128), 0, stream>>>(X, rproj, out_W,
                                                              l1, l2, out_loss);
}